// Net_16982300688706
// MI455X (gfx1250) — compile-verified
//
#include <hip/hip_runtime.h>

typedef __attribute__((ext_vector_type(2))) float v2f;
typedef __attribute__((ext_vector_type(8))) float v8f;

#define IN_DIM 128
#define HID 16

// ---------------- degree / normalization ----------------
__global__ void k_init_deg(float* __restrict__ deg, int n) {
  int i = blockIdx.x * blockDim.x + threadIdx.x;
  if (i < n) deg[i] = 1.0f;  // self-loop contributes 1
}

__global__ void k_count_deg(const int* __restrict__ dst, float* __restrict__ deg, int E) {
  int i = blockIdx.x * blockDim.x + threadIdx.x;
  if (i < E) unsafeAtomicAdd(&deg[dst[i]], 1.0f);  // native global_atomic_add_f32
}

__global__ void k_dinv(float* __restrict__ deg, int n) {
  int i = blockIdx.x * blockDim.x + threadIdx.x;
  if (i < n) deg[i] = rsqrtf(deg[i]);  // deg >= 1 always (self-loop)
}

// ---------------- GEMM1: h = x @ W1  via V_WMMA_F32_16X16X4_F32 ----------------
// One wave computes a 16x16 output tile (16 rows of x, all 16 hidden features).
// A layout (16x4 f32): lane m=lane&15 holds row m; lanes 0-15 K={k,k+1}, lanes 16-31 K={k+2,k+3}.
// B layout (4x16 f32): lane n=lane&15 holds column n; same K split across half-waves.
// D layout: VGPR v -> row (v + 8*(lane>>4)), col = lane&15.
__global__ void k_gemm1_wmma(const float* __restrict__ x,
                             const float* __restrict__ W,
                             float* __restrict__ h, int n_nodes) {
  __shared__ float Ws[IN_DIM * HID];
  for (int i = threadIdx.x; i < IN_DIM * HID; i += blockDim.x) Ws[i] = W[i];
  __syncthreads();

  const int lane  = threadIdx.x & 31;
  const int wave  = (blockIdx.x * blockDim.x + threadIdx.x) >> 5;
  const int row0  = wave * 16;
  if (row0 >= n_nodes) return;  // wave-uniform: EXEC stays all-ones for WMMA

  const int m     = lane & 15;   // A row / B,D column
  const int khalf = lane >> 4;   // upper half-wave handles K+2
  const int ra    = min(row0 + m, n_nodes - 1);  // clamp keeps EXEC full on tail

  v8f acc = {};
#pragma unroll
  for (int k = 0; k < IN_DIM; k += 4) {
    const int k0 = k + 2 * khalf;
    v2f a, b;
    a.x = x[(size_t)ra * IN_DIM + k0];
    a.y = x[(size_t)ra * IN_DIM + k0 + 1];
    b.x = Ws[k0 * HID + m];
    b.y = Ws[(k0 + 1) * HID + m];
    acc = __builtin_amdgcn_wmma_f32_16x16x4_f32(false, a, false, b,
                                                (short)0, acc, false, false);
  }

#pragma unroll
  for (int v = 0; v < 8; ++v) {
    const int r = row0 + v + 8 * khalf;
    if (r < n_nodes) h[(size_t)r * HID + m] = acc[v];
  }
}

// ---------------- GEMM2: h2 = relu(out1) @ W2 (K=16 -> 4 WMMA steps) ----------------
__global__ void k_gemm2_wmma(const float* __restrict__ a_in,  // out1, pre-ReLU [N,16]
                             const float* __restrict__ W,     // [16,16]
                             float* __restrict__ h, int n_nodes) {
  __shared__ float Ws[HID * HID];
  if (threadIdx.x < HID * HID) Ws[threadIdx.x] = W[threadIdx.x];
  __syncthreads();

  const int lane  = threadIdx.x & 31;
  const int wave  = (blockIdx.x * blockDim.x + threadIdx.x) >> 5;
  const int row0  = wave * 16;
  if (row0 >= n_nodes) return;

  const int m     = lane & 15;
  const int khalf = lane >> 4;
  const int ra    = min(row0 + m, n_nodes - 1);

  v8f acc = {};
#pragma unroll
  for (int k = 0; k < HID; k += 4) {
    const int k0 = k + 2 * khalf;
    v2f a, b;
    a.x = fmaxf(a_in[(size_t)ra * HID + k0], 0.0f);      // fused ReLU
    a.y = fmaxf(a_in[(size_t)ra * HID + k0 + 1], 0.0f);
    b.x = Ws[k0 * HID + m];
    b.y = Ws[(k0 + 1) * HID + m];
    acc = __builtin_amdgcn_wmma_f32_16x16x4_f32(false, a, false, b,
                                                (short)0, acc, false, false);
  }

#pragma unroll
  for (int v = 0; v < 8; ++v) {
    const int r = row0 + v + 8 * khalf;
    if (r < n_nodes) h[(size_t)r * HID + m] = acc[v];
  }
}

// ---------------- out[i] = bias + dinv^2 * h[i]   (self-loop term + init) --------
__global__ void k_init_out(const float* __restrict__ h, const float* __restrict__ dinv,
                           const float* __restrict__ bias, float* __restrict__ out,
                           int n_nodes) {
  int i = blockIdx.x * blockDim.x + threadIdx.x;
  if (i >= n_nodes * HID) return;
  int node = i >> 4, f = i & 15;
  float di = dinv[node];
  out[i] = bias[f] + di * di * h[i];
}

// ---------------- edge scatter: 16 lanes per edge, one feature per lane ----------
__global__ void k_scatter(const float* __restrict__ h, const float* __restrict__ dinv,
                          const int* __restrict__ src, const int* __restrict__ dst,
                          float* __restrict__ out, int E) {
  int t = blockIdx.x * blockDim.x + threadIdx.x;
  int e = t >> 4;
  if (e >= E) return;
  int f = t & 15;
  int s = src[e], d = dst[e];
  float w = dinv[s] * dinv[d];
  unsafeAtomicAdd(&out[(size_t)d * HID + f], w * h[(size_t)s * HID + f]);
}

extern "C" void kernel_launch(void* const* d_in, const int* in_sizes, int n_in,
                              void* d_out, int out_size, void* d_ws, size_t ws_size,
                              hipStream_t stream) {
  const float* x  = (const float*)d_in[0];
  const int* src  = (const int*)d_in[1];
  const int* dst  = (const int*)d_in[2];
  const float* W1 = (const float*)d_in[3];
  const float* b1 = (const float*)d_in[4];
  const float* W2 = (const float*)d_in[5];
  const float* b2 = (const float*)d_in[6];
  float* out      = (float*)d_out;

  const int n_nodes = in_sizes[0] / IN_DIM;
  const int E       = in_sizes[1];

  // workspace layout: dinv | h (reused L1/L2) | out1
  char* ws = (char*)d_ws;
  float* dinv = (float*)ws;
  size_t off = ((size_t)n_nodes * sizeof(float) + 255) & ~(size_t)255;
  float* h = (float*)(ws + off);
  off += ((size_t)n_nodes * HID * sizeof(float) + 255) & ~(size_t)255;
  float* out1 = (float*)(ws + off);

  const int T = 256;
  const int nb_nodes = (n_nodes + T - 1) / T;
  const int nb_feats = (n_nodes * HID + T - 1) / T;
  const int nb_edges = (E + T - 1) / T;
  const long long st = (long long)E * HID;
  const int nb_scat  = (int)((st + T - 1) / T);
  const int tiles    = (n_nodes + 15) / 16;
  const int nb_gemm  = (tiles + 7) / 8;  // 8 waves (wave32) per 256-thread block

  // shared symmetric normalization
  k_init_deg<<<nb_nodes, T, 0, stream>>>(dinv, n_nodes);
  k_count_deg<<<nb_edges, T, 0, stream>>>(dst, dinv, E);
  k_dinv<<<nb_nodes, T, 0, stream>>>(dinv, n_nodes);

  // layer 1
  k_gemm1_wmma<<<nb_gemm, T, 0, stream>>>(x, W1, h, n_nodes);
  k_init_out<<<nb_feats, T, 0, stream>>>(h, dinv, b1, out1, n_nodes);
  k_scatter<<<nb_scat, T, 0, stream>>>(h, dinv, src, dst, out1, E);

  // layer 2 (ReLU fused into GEMM2's A-matrix read)
  k_gemm2_wmma<<<nb_gemm, T, 0, stream>>>(out1, W2, h, n_nodes);
  k_init_out<<<nb_feats, T, 0, stream>>>(h, dinv, b2, out, n_nodes);
  k_scatter<<<nb_scat, T, 0, stream>>>(h, dinv, src, dst, out, E);
}